// SegmentationTransformer_33432025432183
// MI455X (gfx1250) — compile-verified
//
#include <hip/hip_runtime.h>

// ---------------------------------------------------------------------------
// CDNA5 (gfx1250) fp32 implementation of the segmentation transformer.
// All GEMM-shaped work goes through V_WMMA_F32_16X16X4_F32 (wave32 WMMA).
// Fragment layouts per CDNA5 ISA 7.12.2:
//   A (16x4 f32, 2 VGPR): lanes 0-15 hold K=k,k+1 ; lanes 16-31 hold K=k+2,k+3
//   B (4x16 f32, 2 VGPR): vgpr v, lanes 0-15 = row k+v ; lanes 16-31 = row k+v+2
//   C/D (16x16 f32, 8 VGPR): vgpr v, lanes 0-15 = M=v ; lanes 16-31 = M=v+8
// ---------------------------------------------------------------------------

typedef float v2f __attribute__((ext_vector_type(2)));
typedef float v8f __attribute__((ext_vector_type(8)));

#define WMMA_F32(a, b, c) \
  __builtin_amdgcn_wmma_f32_16x16x4_f32(false, (a), false, (b), (short)0, (c), false, false)

// ------------------------------ generic GEMM -------------------------------
// C[b][m][n] = act( sum_k A[b][m][k] * B[b][k][n] + bias[n] + R[b][m][n] )
// grid = (N/32, M/64, batches). block = 256 threads (8 waves -> 64x32 tile).
// Requires M%64==0, N%32==0, K%16==0 (true for every call site here).
__global__ void k_gemm(const float* __restrict__ A, const float* __restrict__ B,
                       float* __restrict__ C, const float* __restrict__ bias,
                       const float* __restrict__ R,
                       int M, int N, int K, int lda, int ldb, int ldc, int ldr,
                       long sA, long sB, long sC, long sR,
                       int transB, int act) {
  __shared__ float As[64 * 17];
  __shared__ float Bs[16 * 33];
  const int tid = threadIdx.x, lane = tid & 31, wid = tid >> 5;
  const int wy = wid & 3, wx = wid >> 2, half = lane >> 4, lm = lane & 15;
  const int m0 = blockIdx.y * 64, n0 = blockIdx.x * 32;
  const float* Ab = A + sA * blockIdx.z;
  const float* Bb = B + sB * blockIdx.z;
  float* Cb = C + sC * blockIdx.z;
  const float* Rb = R ? R + sR * blockIdx.z : (const float*)0;

  const int ar = tid >> 2, ak = (tid & 3) * 4;   // A stage: row ar, k ak..ak+3
  const int bk = tid >> 4, bn = (tid & 15) * 2;  // B stage: k bk, n bn..bn+1
  v8f acc = {};
  for (int k0 = 0; k0 < K; k0 += 16) {
    const float* ap = Ab + (long)(m0 + ar) * lda + k0 + ak;
#pragma unroll
    for (int i = 0; i < 4; ++i) As[ar * 17 + ak + i] = ap[i];
#pragma unroll
    for (int j = 0; j < 2; ++j) {
      int n = n0 + bn + j;
      Bs[bk * 33 + bn + j] =
          transB ? Bb[(long)n * ldb + k0 + bk] : Bb[(long)(k0 + bk) * ldb + n];
    }
    __syncthreads();
#pragma unroll
    for (int kk = 0; kk < 16; kk += 4) {
      v2f a, b;
      a.x = As[(wy * 16 + lm) * 17 + kk + 2 * half];
      a.y = As[(wy * 16 + lm) * 17 + kk + 2 * half + 1];
      b.x = Bs[(kk + 2 * half) * 33 + wx * 16 + lm];
      b.y = Bs[(kk + 2 * half + 1) * 33 + wx * 16 + lm];
      acc = WMMA_F32(a, b, acc);
    }
    __syncthreads();
  }
  const int gn = n0 + wx * 16 + lm;
#pragma unroll
  for (int v = 0; v < 8; ++v) {
    int gm = m0 + wy * 16 + v + 8 * half;
    float val = acc[v];
    if (bias) val += bias[gn];
    if (Rb) val += Rb[(long)gm * ldr + gn];
    if (act == 1) val = fmaxf(val, 0.f);
    else if (act == 2) val = 0.5f * val * (1.f + erff(val * 0.70710678118654752f));
    Cb[(long)gm * ldc + gn] = val;
  }
}

// ------------------------------ attention ----------------------------------
// grid = (S/16, HEADS, B), block = 32 (one wave). S=896, HD=32, 768-wide qkv.
__global__ void k_attn(const float* __restrict__ qkv, float* __restrict__ out) {
  const int S = 896;
  __shared__ float P[16 * 900];
  __shared__ float Qs[16 * 33];
  __shared__ float rsum[16];
  const int lane = threadIdx.x, half = lane >> 4, lm = lane & 15;
  const int m0 = blockIdx.x * 16, head = blockIdx.y, b = blockIdx.z;
  const float* base = qkv + (long)b * S * 768;
  const int hq = head * 32;

  for (int i = lane; i < 16 * 32; i += 32) {
    int r = i >> 5, c = i & 31;
    Qs[r * 33 + c] = base[(long)(m0 + r) * 768 + hq + c];
  }
  __syncthreads();

  const float scale = 0.17677669529663689f;  // 1/sqrt(32)
  for (int n0 = 0; n0 < S; n0 += 16) {
    v8f acc = {};
#pragma unroll
    for (int kk = 0; kk < 32; kk += 4) {
      v2f a, bb;
      a.x = Qs[lm * 33 + kk + 2 * half];
      a.y = Qs[lm * 33 + kk + 2 * half + 1];
      bb.x = base[(long)(n0 + lm) * 768 + 256 + hq + kk + 2 * half];
      bb.y = base[(long)(n0 + lm) * 768 + 256 + hq + kk + 2 * half + 1];
      acc = WMMA_F32(a, bb, acc);
    }
#pragma unroll
    for (int v = 0; v < 8; ++v) P[(v + 8 * half) * 900 + n0 + lm] = acc[v] * scale;
  }
  __syncthreads();

  if (lane < 16) {
    float mx = -1e30f;
    for (int j = 0; j < S; ++j) mx = fmaxf(mx, P[lane * 900 + j]);
    float sum = 0.f;
    for (int j = 0; j < S; ++j) {
      float e = expf(P[lane * 900 + j] - mx);
      P[lane * 900 + j] = e;
      sum += e;
    }
    rsum[lane] = 1.f / sum;
  }
  __syncthreads();

  for (int n0 = 0; n0 < 32; n0 += 16) {
    v8f acc = {};
    for (int kk = 0; kk < S; kk += 4) {
      v2f a, bb;
      a.x = P[lm * 900 + kk + 2 * half];
      a.y = P[lm * 900 + kk + 2 * half + 1];
      bb.x = base[(long)(kk + 2 * half) * 768 + 512 + hq + n0 + lm];
      bb.y = base[(long)(kk + 2 * half + 1) * 768 + 512 + hq + n0 + lm];
      acc = WMMA_F32(a, bb, acc);
    }
#pragma unroll
    for (int v = 0; v < 8; ++v) {
      int r = v + 8 * half;
      out[((long)b * S + m0 + r) * 256 + hq + n0 + lm] = acc[v] * rsum[r];
    }
  }
}

// --------------------------- conv3x3 (implicit GEMM) -----------------------
// grid = (ceil(Cout/32), B*H*W/64). K = Cin*9, k = c*9 + ky*3 + kx (OIHW).
__global__ void k_conv3(const float* __restrict__ in, const float* __restrict__ wgt,
                        const float* __restrict__ bias, float* __restrict__ out,
                        int Cin, int Cout, int H, int W) {
  __shared__ float As[64 * 17];
  __shared__ float Bs[16 * 33];
  const int K = Cin * 9;
  const int tid = threadIdx.x, lane = tid & 31, wid = tid >> 5;
  const int wy = wid & 3, wx = wid >> 2, half = lane >> 4, lm = lane & 15;
  const int n0 = blockIdx.x * 32;
  const long m0 = (long)blockIdx.y * 64;
  const int HW = H * W;
  const int ar = tid >> 2, ak4 = (tid & 3) * 4;
  const int bkr = tid >> 4, bnr = (tid & 15) * 2;
  const long m = m0 + ar;
  const int bIdx = (int)(m / HW), hw = (int)(m % HW);
  const int h = hw / W, w = hw % W;
  v8f acc = {};
  for (int k0 = 0; k0 < K; k0 += 16) {
#pragma unroll
    for (int i = 0; i < 4; ++i) {
      int k = k0 + ak4 + i;
      int c = k / 9, r9 = k - 9 * c;
      int ky = r9 / 3, kx = r9 - 3 * ky;
      int ih = h + ky - 1, iw = w + kx - 1;
      float v = 0.f;
      if (ih >= 0 && ih < H && iw >= 0 && iw < W)
        v = in[(((long)bIdx * Cin + c) * H + ih) * W + iw];
      As[ar * 17 + ak4 + i] = v;
    }
#pragma unroll
    for (int j = 0; j < 2; ++j) {
      int n = n0 + bnr + j;
      Bs[bkr * 33 + bnr + j] = (n < Cout) ? wgt[(long)n * K + k0 + bkr] : 0.f;
    }
    __syncthreads();
#pragma unroll
    for (int kk = 0; kk < 16; kk += 4) {
      v2f a, bb;
      a.x = As[(wy * 16 + lm) * 17 + kk + 2 * half];
      a.y = As[(wy * 16 + lm) * 17 + kk + 2 * half + 1];
      bb.x = Bs[(kk + 2 * half) * 33 + wx * 16 + lm];
      bb.y = Bs[(kk + 2 * half + 1) * 33 + wx * 16 + lm];
      acc = WMMA_F32(a, bb, acc);
    }
    __syncthreads();
  }
  const int gn = n0 + wx * 16 + lm;
  if (gn < Cout) {
#pragma unroll
    for (int v = 0; v < 8; ++v) {
      long gm = m0 + wy * 16 + v + 8 * half;
      int ob = (int)(gm / HW), ohw = (int)(gm % HW);
      out[((long)ob * Cout + gn) * HW + ohw] = acc[v] + bias[gn];
    }
  }
}

// ----------------------- transposed conv 2x2 stride 2 ----------------------
// out[b,d,2h+i,2w+j] = sum_c in[b,c,h,w] * wt[c,d,i,j] + bias[d]
// 4 output taps share one A fragment per K step -> 4 WMMAs per step.
__global__ void k_convt2(const float* __restrict__ in, const float* __restrict__ wgt,
                         const float* __restrict__ bias, float* __restrict__ out,
                         int Cc, int Dd, int H, int W) {
  __shared__ float As[64 * 17];
  __shared__ float Bs[4][16 * 33];
  const int tid = threadIdx.x, lane = tid & 31, wid = tid >> 5;
  const int wy = wid & 3, wx = wid >> 2, half = lane >> 4, lm = lane & 15;
  const int n0 = blockIdx.x * 32;
  const long m0 = (long)blockIdx.y * 64;
  const int HW = H * W;
  const int ar = tid >> 2, ak4 = (tid & 3) * 4;
  const int bkr = tid >> 4, bnr = (tid & 15) * 2;
  const long m = m0 + ar;
  const int bIdx = (int)(m / HW), hw = (int)(m % HW);
  v8f acc[4] = {};
  for (int k0 = 0; k0 < Cc; k0 += 16) {
#pragma unroll
    for (int i = 0; i < 4; ++i) {
      int c = k0 + ak4 + i;
      As[ar * 17 + ak4 + i] = in[((long)bIdx * Cc + c) * HW + hw];
    }
#pragma unroll
    for (int j = 0; j < 2; ++j) {
      int n = n0 + bnr + j;
#pragma unroll
      for (int ij = 0; ij < 4; ++ij)
        Bs[ij][bkr * 33 + bnr + j] = wgt[((long)(k0 + bkr) * Dd + n) * 4 + ij];
    }
    __syncthreads();
#pragma unroll
    for (int kk = 0; kk < 16; kk += 4) {
      v2f a;
      a.x = As[(wy * 16 + lm) * 17 + kk + 2 * half];
      a.y = As[(wy * 16 + lm) * 17 + kk + 2 * half + 1];
      const int bo = wx * 16 + lm;
#pragma unroll
      for (int ij = 0; ij < 4; ++ij) {
        v2f bb;
        bb.x = Bs[ij][(kk + 2 * half) * 33 + bo];
        bb.y = Bs[ij][(kk + 2 * half + 1) * 33 + bo];
        acc[ij] = WMMA_F32(a, bb, acc[ij]);
      }
    }
    __syncthreads();
  }
  const int gn = n0 + wx * 16 + lm;
  const float bv = bias[gn];
#pragma unroll
  for (int v = 0; v < 8; ++v) {
    long gm = m0 + wy * 16 + v + 8 * half;
    int ob = (int)(gm / HW), ohw = (int)(gm % HW);
    int h = ohw / W, w = ohw % W;
    long rb = ((long)ob * Dd + gn) * (2 * H);
    out[(rb + 2 * h + 0) * (2 * W) + 2 * w + 0] = acc[0][v] + bv;
    out[(rb + 2 * h + 0) * (2 * W) + 2 * w + 1] = acc[1][v] + bv;
    out[(rb + 2 * h + 1) * (2 * W) + 2 * w + 0] = acc[2][v] + bv;
    out[(rb + 2 * h + 1) * (2 * W) + 2 * w + 1] = acc[3][v] + bv;
  }
}

// --------------------------- small helper kernels --------------------------
__global__ void k_pos(const int* __restrict__ pc, const float* __restrict__ w1,
                      const float* __restrict__ b1, const float* __restrict__ w2,
                      const float* __restrict__ b2, float* __restrict__ pos) {
  __shared__ float hs[256];
  const int r = blockIdx.x, t = threadIdx.x;
  float p0 = (float)pc[2 * r], p1 = (float)pc[2 * r + 1];
  float hv = p0 * w1[t] + p1 * w1[256 + t] + b1[t];
  hs[t] = fmaxf(hv, 0.f);
  __syncthreads();
  float acc = b2[t];
  for (int j = 0; j < 256; ++j) acc += hs[j] * w2[j * 256 + t];
  pos[(long)r * 256 + t] = acc;
}

__global__ void k_cls(const float* __restrict__ cls, float* __restrict__ x) {
  int idx = blockIdx.x * 256 + threadIdx.x;  // B*256*256
  int d = idx & 255, i = (idx >> 8) & 255, b = idx >> 16;
  x[((long)b * 896 + 640 + i) * 256 + d] = cls[i * 256 + d];
}

__global__ void k_ln(const float* __restrict__ in, float* __restrict__ out,
                     const float* __restrict__ g, const float* __restrict__ b) {
  __shared__ float red[256];
  const long r = blockIdx.x;
  const int t = threadIdx.x;
  float v = in[r * 256 + t];
  red[t] = v;
  __syncthreads();
  for (int s = 128; s > 0; s >>= 1) { if (t < s) red[t] += red[t + s]; __syncthreads(); }
  float mean = red[0] * (1.f / 256.f);
  __syncthreads();
  float d = v - mean;
  red[t] = d * d;
  __syncthreads();
  for (int s = 128; s > 0; s >>= 1) { if (t < s) red[t] += red[t + s]; __syncthreads(); }
  float var = red[0] * (1.f / 256.f);
  out[r * 256 + t] = d * rsqrtf(var + 1e-5f) * g[t] + b[t];
}

__global__ void k_l2(float* __restrict__ x) {
  __shared__ float red[256];
  const long r = blockIdx.x;
  const int t = threadIdx.x;
  float v = x[r * 256 + t];
  red[t] = v * v;
  __syncthreads();
  for (int s = 128; s > 0; s >>= 1) { if (t < s) red[t] += red[t + s]; __syncthreads(); }
  x[r * 256 + t] = v * rsqrtf(red[0]);
}

__global__ void k_build_inv(const int* __restrict__ pc, int* __restrict__ inv) {
  const int p = threadIdx.x;
  if (p < 640) {
    int s = pc[2 * p], coord = pc[2 * p + 1];
    int gs = 16 << s;
    int k = 1 << (1 - s);
    int r = (coord / gs) * k, c = (coord % gs) * k;
    for (int i = 0; i < k; ++i)
      for (int j = 0; j < k; ++j) inv[(r + i) * 32 + c + j] = p;
  }
}

__global__ void k_gather(const float* __restrict__ masks, const int* __restrict__ inv,
                         float* __restrict__ img) {
  int idx = blockIdx.x * 256 + threadIdx.x;  // B*256*1024
  int cell = idx & 1023, d = (idx >> 10) & 255, b = idx >> 18;
  img[idx] = masks[((long)b * 640 + inv[cell]) * 256 + d];
}

__global__ void k_inorm_leaky(float* __restrict__ x, int HW) {
  // grid = (C=256, B); instance-norm over HW then leaky ReLU, in place.
  const int c = blockIdx.x, b = blockIdx.y, t = threadIdx.x;
  float* p = x + ((long)b * gridDim.x + c) * HW;
  __shared__ float r1[256], r2[256];
  float s = 0.f, s2 = 0.f;
  for (int i = t; i < HW; i += 256) { float v = p[i]; s += v; s2 += v * v; }
  r1[t] = s; r2[t] = s2;
  __syncthreads();
  for (int st = 128; st > 0; st >>= 1) {
    if (t < st) { r1[t] += r1[t + st]; r2[t] += r2[t + st]; }
    __syncthreads();
  }
  float mean = r1[0] / (float)HW;
  float var = r2[0] / (float)HW - mean * mean;
  float rinv = rsqrtf(var + 1e-5f);
  for (int i = t; i < HW; i += 256) {
    float v = (p[i] - mean) * rinv;
    p[i] = v >= 0.f ? v : 0.01f * v;
  }
}

// ------------------------------- launcher ----------------------------------
extern "C" void kernel_launch(void* const* d_in, const int* in_sizes, int n_in,
                              void* d_out, int out_size, void* d_ws, size_t ws_size,
                              hipStream_t stream) {
  const float* x_in = (const float*)d_in[0];
  const int* pcode = (const int*)d_in[1];
  const float* proj_dec_w = (const float*)d_in[2];
  const float* proj_dec_b = (const float*)d_in[3];
  const float* pos_w1 = (const float*)d_in[4];
  const float* pos_b1 = (const float*)d_in[5];
  const float* pos_w2 = (const float*)d_in[6];
  const float* pos_b2 = (const float*)d_in[7];
  const float* cls_emb = (const float*)d_in[8];
  const float* ln1_g = (const float*)d_in[9];
  const float* ln1_b = (const float*)d_in[10];
  const float* qkv_w = (const float*)d_in[11];
  const float* qkv_b = (const float*)d_in[12];
  const float* attn_w = (const float*)d_in[13];
  const float* attn_b = (const float*)d_in[14];
  const float* ln2_g = (const float*)d_in[15];
  const float* ln2_b = (const float*)d_in[16];
  const float* mlp_w1 = (const float*)d_in[17];
  const float* mlp_b1 = (const float*)d_in[18];
  const float* mlp_w2 = (const float*)d_in[19];
  const float* mlp_b2 = (const float*)d_in[20];
  const float* dec_g = (const float*)d_in[21];
  const float* dec_b = (const float*)d_in[22];
  const float* proj_patch = (const float*)d_in[23];
  const float* proj_classes = (const float*)d_in[24];
  const float* mask_g = (const float*)d_in[25];
  const float* mask_b = (const float*)d_in[26];
  const float* up_conv_w = (const float*)d_in[27];
  const float* up_conv_b = (const float*)d_in[28];
  const float* up_convt_w = (const float*)d_in[29];
  const float* up_convt_b = (const float*)d_in[30];
  const float* out_conv_w = (const float*)d_in[31];
  const float* out_conv_b = (const float*)d_in[32];

  // ---- workspace carve (sequential, 256B aligned) ----
  char* ws = (char*)d_ws;
  size_t off = 0;
  auto carve = [&](size_t bytes) -> void* {
    void* p = ws + off;
    off += (bytes + 255) & ~(size_t)255;
    return p;
  };
  int* inv = (int*)carve(1024 * sizeof(int));
  float* masks = (float*)carve((size_t)2 * 640 * 256 * 4);     // B,640,256
  float* posb = (float*)carve((size_t)640 * 256 * 4);
  float* xbuf = (float*)carve((size_t)2 * 896 * 256 * 4);      // B,896,256
  float* hbuf = (float*)carve((size_t)2 * 896 * 256 * 4);
  float* qkvb = (float*)carve((size_t)2 * 896 * 768 * 4);
  float* attno = (float*)carve((size_t)2 * 896 * 256 * 4);
  float* mlph = (float*)carve((size_t)2 * 896 * 1024 * 4);
  float* pproj = (float*)carve((size_t)2 * 640 * 256 * 4);
  float* cproj = (float*)carve((size_t)2 * 256 * 256 * 4);
  float* bufB = (float*)carve((size_t)2 * 256 * 128 * 128 * 4);  // conv scratch
  float* bufA = (float*)carve((size_t)2 * 256 * 256 * 256 * 4);  // conv scratch
  (void)ws_size; (void)in_sizes; (void)n_in; (void)out_size;

  // 1) positional MLP: pos = relu(pc@w1+b1)@w2+b2
  k_pos<<<640, 256, 0, stream>>>(pcode, pos_w1, pos_b1, pos_w2, pos_b2, posb);

  // 2) embed: x[b,0:640,:] = x_in@W + b + pos   (batched, M=640,N=256,K=512)
  k_gemm<<<dim3(8, 10, 2), 256, 0, stream>>>(
      x_in, proj_dec_w, xbuf, proj_dec_b, posb,
      640, 256, 512, 512, 256, 256, 256,
      (long)640 * 512, 0, (long)896 * 256, 0, 0, 0);

  // 3) cls rows
  k_cls<<<512, 256, 0, stream>>>(cls_emb, xbuf);

  // 4) transformer blocks
  for (int l = 0; l < 2; ++l) {
    k_ln<<<1792, 256, 0, stream>>>(xbuf, hbuf, ln1_g + l * 256, ln1_b + l * 256);
    // qkv: (1792,256)@(256,768)
    k_gemm<<<dim3(24, 28, 1), 256, 0, stream>>>(
        hbuf, qkv_w + (size_t)l * 256 * 768, qkvb, qkv_b + l * 768, (const float*)0,
        1792, 768, 256, 256, 768, 768, 0, 0, 0, 0, 0, 0, 0);
    k_attn<<<dim3(56, 8, 2), 32, 0, stream>>>(qkvb, attno);
    // out proj + residual into x
    k_gemm<<<dim3(8, 28, 1), 256, 0, stream>>>(
        attno, attn_w + (size_t)l * 256 * 256, xbuf, attn_b + l * 256, xbuf,
        1792, 256, 256, 256, 256, 256, 256, 0, 0, 0, 0, 0, 0);
    k_ln<<<1792, 256, 0, stream>>>(xbuf, hbuf, ln2_g + l * 256, ln2_b + l * 256);
    // mlp1 + exact GELU
    k_gemm<<<dim3(32, 28, 1), 256, 0, stream>>>(
        hbuf, mlp_w1 + (size_t)l * 256 * 1024, mlph, mlp_b1 + l * 1024, (const float*)0,
        1792, 1024, 256, 256, 1024, 1024, 0, 0, 0, 0, 0, 0, 2);
    // mlp2 + residual into x
    k_gemm<<<dim3(8, 28, 1), 256, 0, stream>>>(
        mlph, mlp_w2 + (size_t)l * 1024 * 256, xbuf, mlp_b2 + l * 256, xbuf,
        1792, 256, 1024, 1024, 256, 256, 256, 0, 0, 0, 0, 0, 0);
  }

  // 5) decoder LN
  k_ln<<<1792, 256, 0, stream>>>(xbuf, hbuf, dec_g, dec_b);

  // 6) projections (batched over B)
  k_gemm<<<dim3(8, 10, 2), 256, 0, stream>>>(
      hbuf, proj_patch, pproj, (const float*)0, (const float*)0,
      640, 256, 256, 256, 256, 256, 0,
      (long)896 * 256, 0, (long)640 * 256, 0, 0, 0);
  k_gemm<<<dim3(8, 4, 2), 256, 0, stream>>>(
      hbuf + (size_t)640 * 256, proj_classes, cproj, (const float*)0, (const float*)0,
      256, 256, 256, 256, 256, 256, 0,
      (long)896 * 256, 0, (long)256 * 256, 0, 0, 0);

  // 7) L2 normalize rows
  k_l2<<<1280, 256, 0, stream>>>(pproj);
  k_l2<<<512, 256, 0, stream>>>(cproj);

  // 8) masks = patches @ cls^T  (batched, transB), then LN
  k_gemm<<<dim3(8, 10, 2), 256, 0, stream>>>(
      pproj, cproj, masks, (const float*)0, (const float*)0,
      640, 256, 256, 256, 256, 256, 0,
      (long)640 * 256, (long)256 * 256, (long)640 * 256, 0, 1, 0);
  k_ln<<<1280, 256, 0, stream>>>(masks, masks, mask_g, mask_b);

  // 9) scatter to 32x32 grid
  k_build_inv<<<1, 640, 0, stream>>>(pcode, inv);
  k_gather<<<2048, 256, 0, stream>>>(masks, inv, bufA);

  // 10) upsampling stages
  int Hs = 32;
  for (int i = 0; i < 3; ++i) {
    int HW = Hs * Hs;
    k_conv3<<<dim3(8, (2 * HW) / 64), 256, 0, stream>>>(
        bufA, up_conv_w + (size_t)i * 256 * 2304, up_conv_b + i * 256, bufB,
        256, 256, Hs, Hs);
    k_inorm_leaky<<<dim3(256, 2), 256, 0, stream>>>(bufB, HW);
    k_convt2<<<dim3(8, (2 * HW) / 64), 256, 0, stream>>>(
        bufB, up_convt_w + (size_t)i * 256 * 256 * 4, up_convt_b + i * 256, bufA,
        256, 256, Hs, Hs);
    k_inorm_leaky<<<dim3(256, 2), 256, 0, stream>>>(bufA, 4 * HW);
    Hs *= 2;
  }

  // 11) output conv (Cout=3, guarded N tile), straight into d_out
  k_conv3<<<dim3(1, (2 * 256 * 256) / 64), 256, 0, stream>>>(
      bufA, out_conv_w, out_conv_b, (float*)d_out, 256, 3, 256, 256);
}